// ZLangevinBlock_14319420965444
// MI455X (gfx1250) — compile-verified
//
#include <hip/hip_runtime.h>
#include <math.h>

// Problem constants (B=8, S=2048, D=768, 5 steps)
#define B_DIM   8
#define S_LEN   2048
#define D_DIM   768
#define NTOK    (B_DIM * S_LEN)      // 16384
#define NSTEPS  5
#define STEP_SZ 0.01f
#define NOISE_SCALE 0.0447213595499958f   // sqrt(0.01)*sqrt(2*0.1)
#define LN_EPS  1e-5f
#define SCHUNKS 16                   // S split for ctx-mean WMMA pass (concurrency)
#define CHUNK_S (S_LEN / SCHUNKS)    // 128 rows -> 32 WMMA iters of K=4

typedef __attribute__((ext_vector_type(2))) float v2f;
typedef __attribute__((ext_vector_type(8))) float v8f;

// ---------------------------------------------------------------------------
// Wave32 all-reduce sum (fixed butterfly order -> deterministic)
// ---------------------------------------------------------------------------
__device__ __forceinline__ float wave_allreduce(float v) {
    #pragma unroll
    for (int m = 16; m > 0; m >>= 1) v += __shfl_xor(v, m, 32);
    return v;
}

// ---------------------------------------------------------------------------
// Kernel A: column sums of context over an S-chunk via V_WMMA_F32_16X16X4_F32.
// One wave per (chunk, b, 16-wide d-tile). A = all-ones 16x4, so
// C[m][n] = sum_k B[k][n] regardless of the B-matrix k-axis VGPR layout
// (summation over k is symmetric). C/D layout (ISA 7.12.2): VGPR0,
// lanes 0-15 hold row M=0, N=lane.
// grid = SCHUNKS * B_DIM * (D_DIM/16) = 6144 blocks of 32 threads.
// ---------------------------------------------------------------------------
__global__ void ctx_partial_kernel(const float* __restrict__ ctx,
                                   float* __restrict__ part) {
    const int lane  = threadIdx.x;            // full wave, EXEC all ones
    const int wid   = blockIdx.x;
    const int chunk = wid / (B_DIM * (D_DIM / 16));
    const int rem   = wid % (B_DIM * (D_DIM / 16));
    const int b     = rem / (D_DIM / 16);
    const int dt    = rem % (D_DIM / 16);
    const int d0    = dt * 16;
    const int half  = lane >> 4;              // which pair of s-rows this lane feeds
    const int n     = lane & 15;              // column within the 16-wide d tile

    // lane reads rows (s + 2*half + {0,1}) of this b
    const float* p = ctx + ((size_t)b * S_LEN + (size_t)chunk * CHUNK_S + 2 * half) * D_DIM
                         + d0 + n;

    v2f a; a[0] = 1.0f; a[1] = 1.0f;          // all-ones A tile
    v8f c = {};                                // zero accumulator

    for (int it = 0; it < CHUNK_S / 4; ++it) {
        v2f bv;
        bv[0] = p[0];
        bv[1] = p[D_DIM];
        c = __builtin_amdgcn_wmma_f32_16x16x4_f32(
                /*neg_a=*/false, a, /*neg_b=*/false, bv,
                /*c_mod=*/(short)0, c, /*reuse_a=*/false, /*reuse_b=*/false);
        p += 4 * D_DIM;
    }

    if (lane < 16) {
        // row M=0 of C: VGPR0, lanes 0-15, N = lane
        part[(size_t)chunk * (B_DIM * D_DIM) + b * D_DIM + d0 + lane] = c[0];
    }
}

// Combine SCHUNKS partials -> ctx_mean (deterministic fixed order)
__global__ void ctx_combine_kernel(const float* __restrict__ part,
                                   float* __restrict__ ctxmean) {
    int i = blockIdx.x * blockDim.x + threadIdx.x;
    if (i < B_DIM * D_DIM) {
        float s = 0.0f;
        #pragma unroll
        for (int ch = 0; ch < SCHUNKS; ++ch) s += part[ch * (B_DIM * D_DIM) + i];
        ctxmean[i] = s * (1.0f / (float)S_LEN);
    }
}

// ---------------------------------------------------------------------------
// Kernel B: fused 5-step Langevin + LayerNorm. One wave per token; each lane
// holds 24 floats of the row (float4 x 6, d = 128*j + 4*lane + c). x stays in
// registers across all steps: x read once, noise streamed, out written once.
// grid = 2048 blocks x 256 threads (8 waves/block).
// ---------------------------------------------------------------------------
__global__ __launch_bounds__(256) void langevin_kernel(
    const float* __restrict__ x_in, const float* __restrict__ noise,
    const float* __restrict__ w_energy, const float* __restrict__ ctxmean,
    const float* __restrict__ gamma, const float* __restrict__ beta,
    float* __restrict__ out, float* __restrict__ epart)
{
    const int wave = threadIdx.x >> 5;
    const int lane = threadIdx.x & 31;
    const int t    = blockIdx.x * 8 + wave;       // token id, 0..16383
    const int b    = t >> 11;                     // S = 2048
    const size_t base = (size_t)t * D_DIM;
    const int off  = 4 * lane;

    float xr[24], cm[24], wr[24];
    #pragma unroll
    for (int j = 0; j < 6; ++j) {
        float4 v;
        v = *(const float4*)(x_in + base + 128 * j + off);
        xr[4*j+0]=v.x; xr[4*j+1]=v.y; xr[4*j+2]=v.z; xr[4*j+3]=v.w;
        v = *(const float4*)(ctxmean + (size_t)b * D_DIM + 128 * j + off);
        cm[4*j+0]=v.x; cm[4*j+1]=v.y; cm[4*j+2]=v.z; cm[4*j+3]=v.w;
        v = *(const float4*)(w_energy + 128 * j + off);
        wr[4*j+0]=v.x; wr[4*j+1]=v.y; wr[4*j+2]=v.z; wr[4*j+3]=v.w;
    }

    #pragma unroll
    for (int step = 0; step < NSTEPS; ++step) {
        // local sum of squared diff
        float ss = 0.0f;
        #pragma unroll
        for (int k = 0; k < 24; ++k) {
            float d = xr[k] - cm[k];
            ss = fmaf(d, d, ss);
        }
        // issue this step's noise loads early to overlap with the reduction
        float nr[24];
        const float* np = noise + (size_t)step * (size_t)NTOK * D_DIM + base;
        #pragma unroll
        for (int j = 0; j < 6; ++j) {
            float4 v = *(const float4*)(np + 128 * j + off);
            nr[4*j+0]=v.x; nr[4*j+1]=v.y; nr[4*j+2]=v.z; nr[4*j+3]=v.w;
        }
        // prefetch next step's noise lines (global_prefetch_b8) while we reduce
        if (step + 1 < NSTEPS) {
            const float* pf = noise + (size_t)(step + 1) * (size_t)NTOK * D_DIM + base;
            #pragma unroll
            for (int j = 0; j < 6; ++j)
                __builtin_prefetch(pf + 128 * j + off, 0, 3);
        }
        ss = wave_allreduce(ss);
        float dist = sqrtf(ss);
        float inv  = 1.0f / dist;

        if (step == NSTEPS - 1) {
            // energy at the start of the last step: dot(x, w) + dist per token
            float dot = 0.0f;
            #pragma unroll
            for (int k = 0; k < 24; ++k) dot = fmaf(xr[k], wr[k], dot);
            dot = wave_allreduce(dot);
            if (lane == 0) epart[t] = dot + dist;
        }

        #pragma unroll
        for (int k = 0; k < 24; ++k) {
            float d = xr[k] - cm[k];
            xr[k] = xr[k] - STEP_SZ * fmaf(d, inv, wr[k]) + NOISE_SCALE * nr[k];
        }
    }

    // LayerNorm over D = 768
    float s1 = 0.0f;
    #pragma unroll
    for (int k = 0; k < 24; ++k) s1 += xr[k];
    s1 = wave_allreduce(s1);
    const float mu = s1 * (1.0f / (float)D_DIM);

    float s2 = 0.0f;
    #pragma unroll
    for (int k = 0; k < 24; ++k) {
        float d = xr[k] - mu;
        s2 = fmaf(d, d, s2);
    }
    s2 = wave_allreduce(s2);
    const float rstd = rsqrtf(s2 * (1.0f / (float)D_DIM) + LN_EPS);

    #pragma unroll
    for (int j = 0; j < 6; ++j) {
        float4 g = *(const float4*)(gamma + 128 * j + off);
        float4 be = *(const float4*)(beta + 128 * j + off);
        float4 o;
        o.x = (xr[4*j+0] - mu) * rstd * g.x + be.x;
        o.y = (xr[4*j+1] - mu) * rstd * g.y + be.y;
        o.z = (xr[4*j+2] - mu) * rstd * g.z + be.z;
        o.w = (xr[4*j+3] - mu) * rstd * g.w + be.w;
        *(float4*)(out + base + 128 * j + off) = o;
    }
}

// ---------------------------------------------------------------------------
// Kernel C: deterministic reduction of 16384 per-token energy partials.
// ---------------------------------------------------------------------------
__global__ void finalize_kernel(const float* __restrict__ epart,
                                const float* __restrict__ b_energy,
                                float* __restrict__ out_scalar) {
    __shared__ float sm[256];
    const int tid = threadIdx.x;
    float s = 0.0f;
    for (int i = tid; i < NTOK; i += 256) s += epart[i];   // fixed order
    sm[tid] = s;
    __syncthreads();
    #pragma unroll
    for (int stride = 128; stride > 0; stride >>= 1) {
        if (tid < stride) sm[tid] += sm[tid + stride];
        __syncthreads();
    }
    if (tid == 0) out_scalar[0] = sm[0] + b_energy[0] * (float)NTOK;
}

// ---------------------------------------------------------------------------
// Launch
// inputs: 0:x 1:context 2:w_energy 3:b_energy 4:ln_gamma 5:ln_beta 6:noise
// d_out: x_final (12582912 floats) then delta_E (1 float)
// ws (floats): [0,6144) ctx_mean | [6144,104448) chunk partials | [104448,120832) energy partials
// total ~483 KB
// ---------------------------------------------------------------------------
extern "C" void kernel_launch(void* const* d_in, const int* in_sizes, int n_in,
                              void* d_out, int out_size, void* d_ws, size_t ws_size,
                              hipStream_t stream) {
    (void)in_sizes; (void)n_in; (void)out_size; (void)ws_size;
    const float* x        = (const float*)d_in[0];
    const float* context  = (const float*)d_in[1];
    const float* w_energy = (const float*)d_in[2];
    const float* b_energy = (const float*)d_in[3];
    const float* ln_gamma = (const float*)d_in[4];
    const float* ln_beta  = (const float*)d_in[5];
    const float* noise    = (const float*)d_in[6];

    float* out = (float*)d_out;
    float* wsf = (float*)d_ws;
    float* ctxmean = wsf;                            // 6144
    float* part    = wsf + B_DIM * D_DIM;            // 16 * 6144
    float* epart   = part + SCHUNKS * B_DIM * D_DIM; // 16384

    // 1) context column sums per S-chunk (WMMA), 6144 single-wave blocks
    ctx_partial_kernel<<<SCHUNKS * B_DIM * (D_DIM / 16), 32, 0, stream>>>(context, part);
    // 2) combine chunks -> ctx_mean
    ctx_combine_kernel<<<(B_DIM * D_DIM + 255) / 256, 256, 0, stream>>>(part, ctxmean);
    // 3) fused 5-step Langevin + LayerNorm, one wave per token
    langevin_kernel<<<NTOK / 8, 256, 0, stream>>>(x, noise, w_energy, ctxmean,
                                                  ln_gamma, ln_beta, out, epart);
    // 4) deterministic energy reduction + bias term
    finalize_kernel<<<1, 256, 0, stream>>>(epart, b_energy, out + (size_t)NTOK * D_DIM);
}